// Encoder_55293408969078
// MI455X (gfx1250) — compile-verified
//
#include <hip/hip_runtime.h>
#include <stdint.h>

// Problem constants (from reference): B=8, L=512, T=4096, D=512
#define B_ 8
#define L_ 512
#define T_ 4096
#define D_ 512

#define TT 256   // t-tile per block (threads per block)
#define DC 64    // d rows per block
#define LCAP 128 // max l-window columns staged in LDS per TDM load

typedef unsigned int u32x4 __attribute__((ext_vector_type(4)));
typedef int i32x8 __attribute__((ext_vector_type(8)));
typedef int i32x4 __attribute__((ext_vector_type(4)));

static __device__ __forceinline__ unsigned rfl(unsigned v) {
  return (unsigned)__builtin_amdgcn_readfirstlane((int)v);
}

// ---------------------------------------------------------------------------
// Expand via segment gather. The duration-cumsum is monotone, so each block's
// t-tile maps onto a CONTIGUOUS l-span of encodings: the TDM stages that
// (DC x span) 2D tile into LDS in async DMAs of up to LCAP columns; lanes then scatter-read LDS
// and store coalesced + non-temporal along t (output is write-once, 64 MB).
// ---------------------------------------------------------------------------
__global__ void __launch_bounds__(TT)
expand_kernel(const float* __restrict__ enc,
              const int* __restrict__ cums,
              float* __restrict__ out) {
  const int b  = blockIdx.z;
  const int d0 = blockIdx.y * DC;
  const int t0 = blockIdx.x * TT;
  const int tid = threadIdx.x;
  const int t = t0 + tid;

  __shared__ int   scums[L_ + 1];
  __shared__ int   sminmax[2];
  __shared__ float tile[DC * LCAP]; // 32 KB staging tile

  for (int i = tid; i < L_ + 1; i += TT) scums[i] = cums[b * (L_ + 1) + i];
  if (tid == 0) { sminmax[0] = L_; sminmax[1] = -1; }
  __syncthreads();

  const int total = scums[L_];
  int lidx = -1;
  if (t < total) {
    // maximal l with scums[l] <= t  (then scums[l+1] > t is guaranteed)
    int lo = 0, hi = L_ - 1;
    while (lo < hi) {
      int mid = (lo + hi + 1) >> 1;
      if (scums[mid] <= t) lo = mid; else hi = mid - 1;
    }
    lidx = lo;
    atomicMin(&sminmax[0], lidx);
    atomicMax(&sminmax[1], lidx);
  }
  __syncthreads();

  const int lmin = sminmax[0];
  const int lmax = sminmax[1];
  const size_t outBase = ((size_t)b * D_ + (size_t)d0) * (size_t)T_ + (size_t)t;

  if (lmax >= 0) {
    for (int lbase = lmin; lbase <= lmax; lbase += LCAP) {
      const int tw = min(LCAP, lmax + 1 - lbase); // tile width (columns)

      if (tid < 32) {
        // ---- Build Tensor DMA Descriptor (D#) for a 2D tile ----
        // dim0 = l (contiguous, stride 1), dim1 = d (stride = L_ elements)
        const unsigned ldsAddr = (unsigned)(uintptr_t)(&tile[0]);
        const unsigned long long ga =
            (unsigned long long)(uintptr_t)enc +
            4ull * (unsigned long long)(((size_t)b * D_ + (size_t)d0) * L_ +
                                        (size_t)lbase);
        const int td0 = L_ - lbase; // remaining columns (OOB guard)
        const int td1 = D_ - d0;    // remaining rows    (OOB guard)

        // Group 0: count=1 | lds_addr | global_addr[56:0] | type=2
        const unsigned g0w0 = 1u;
        const unsigned g0w1 = rfl(ldsAddr);
        const unsigned g0w2 = rfl((unsigned)ga);
        const unsigned g0w3 = rfl(((unsigned)(ga >> 32) & 0x01FFFFFFu) | (2u << 30));
        // Group 1: data_size=4B (code 2); tensor_dim0/1, tile_dim0/1, dim0 stride
        const unsigned g1w0 = 2u << 16;                                   // data_size
        const unsigned g1w1 = rfl(((unsigned)td0 & 0xFFFFu) << 16);        // td0[15:0]
        const unsigned g1w2 = rfl(((unsigned)td0 >> 16) |
                                  (((unsigned)td1 & 0xFFFFu) << 16));      // td0[31:16]|td1[15:0]
        const unsigned g1w3 = rfl(((unsigned)td1 >> 16) |
                                  ((unsigned)tw << 16));                   // td1[31:16]|tile_dim0
        const unsigned g1w4 = (unsigned)DC;                                // tile_dim1 (tile_dim2=0)
        const unsigned g1w5 = (unsigned)L_;                                // tensor_dim0_stride

        u32x4 g0 = {g0w0, g0w1, g0w2, g0w3};
        i32x8 g1 = {(int)g1w0, (int)g1w1, (int)g1w2, (int)g1w3,
                    (int)g1w4, (int)g1w5, 0, 0};
        i32x4 gz4 = {0, 0, 0, 0};
        i32x8 gz8 = {0, 0, 0, 0, 0, 0, 0, 0};
        __builtin_amdgcn_tensor_load_to_lds(g0, g1, gz4, gz4, gz8, 0);
        __builtin_amdgcn_s_wait_tensorcnt(0);
      }
      __syncthreads();

      if (lidx >= lbase && lidx < lbase + tw) {
        const int c = lidx - lbase;
        #pragma unroll 4
        for (int dd = 0; dd < DC; ++dd) {
          // write-once streaming output: non-temporal to spare L2 for enc
          __builtin_nontemporal_store(tile[dd * tw + c],
                                      &out[outBase + (size_t)dd * T_]);
        }
      }
      __syncthreads();
    }
  }

  if (lidx < 0) {
    #pragma unroll 4
    for (int dd = 0; dd < DC; ++dd) {
      __builtin_nontemporal_store(0.0f, &out[outBase + (size_t)dd * T_]);
    }
  }
}

// ---------------------------------------------------------------------------
// Per-batch duration rounding + inclusive scan.
// Emits: repeats (f32) -> rep_out, latent_lengths (f32) -> len_out,
//        cums (int32, L_+1 per batch) -> d_ws scratch.
// ---------------------------------------------------------------------------
__global__ void dur_scan_kernel(const float* __restrict__ dur,
                                float* __restrict__ rep_out,
                                float* __restrict__ len_out,
                                int* __restrict__ cums) {
  const int b = blockIdx.x;
  const int l = threadIdx.x; // 512 threads
  const int r = (int)floorf(dur[b * L_ + l] + 0.5f);
  rep_out[b * L_ + l] = (float)r;

  __shared__ int s[L_];
  s[l] = r;
  __syncthreads();
  // Hillis-Steele inclusive scan over 512 elements
  for (int off = 1; off < L_; off <<= 1) {
    int v = (l >= off) ? s[l - off] : 0;
    __syncthreads();
    s[l] += v;
    __syncthreads();
  }
  if (l == 0) cums[b * (L_ + 1)] = 0;
  cums[b * (L_ + 1) + l + 1] = s[l];
  if (l == L_ - 1) len_out[b] = (float)s[l];
}

// ---------------------------------------------------------------------------
// Launch: d_in[0]=encodings (B,D,L) f32, d_in[1]=durations (B,L) f32,
// d_in[2]=max_length (scalar, ignored; T_ hardcoded).
// d_out = [expanded (B*D*T) | repeats (B*L) | latent_lengths (B)] as f32.
// ---------------------------------------------------------------------------
extern "C" void kernel_launch(void* const* d_in, const int* in_sizes, int n_in,
                              void* d_out, int out_size, void* d_ws, size_t ws_size,
                              hipStream_t stream) {
  (void)in_sizes; (void)n_in; (void)out_size; (void)ws_size;
  const float* enc = (const float*)d_in[0];
  const float* dur = (const float*)d_in[1];

  float* out = (float*)d_out;
  float* rep = out + (size_t)B_ * D_ * T_;
  float* len = rep + (size_t)B_ * L_;
  int* cums = (int*)d_ws; // B_*(L_+1) ints

  dur_scan_kernel<<<dim3(B_), dim3(L_), 0, stream>>>(dur, rep, len, cums);
  expand_kernel<<<dim3(T_ / TT, D_ / DC, B_), dim3(TT), 0, stream>>>(enc, cums, out);
}